// SidedDistance_42872363549144
// MI455X (gfx1250) — compile-verified
//
#include <hip/hip_runtime.h>

typedef float v2f __attribute__((ext_vector_type(2)));
typedef float v4f __attribute__((ext_vector_type(4)));
typedef float v8f __attribute__((ext_vector_type(8)));

#define NPTS          16384
#define CHUNK_PTS     4096          // 48KB LDS (SoA x/y/z) per chunk
#define WAVES_PER_BLK 8
#define BLK_THREADS   256           // 8 wave32 per block (one WGP, 2x over 4 SIMD32)

__global__ __launch_bounds__(BLK_THREADS)
void sided_nn_wmma_kernel(const float* __restrict__ S1,
                          const float* __restrict__ S2,
                          float* __restrict__ out) {
    // SoA candidate chunk: [x | y | z], each CHUNK_PTS floats
    __shared__ float sP[3 * CHUNK_PTS];

    const unsigned tid  = threadIdx.x;
    const unsigned lane = tid & 31u;          // wave32
    const unsigned wave = tid >> 5;
    const unsigned col  = lane & 15u;         // candidate column within 16-tile
    const bool     hi   = lane >= 16u;        // high half-wave (K=2,3 / rows 8..15)

    const unsigned tilesPerBatch  = NPTS / 16;                    // 1024
    const unsigned blocksPerBatch = tilesPerBatch / WAVES_PER_BLK; // 128
    const unsigned b    = blockIdx.x / blocksPerBatch;
    const unsigned tile = (blockIdx.x % blocksPerBatch) * WAVES_PER_BLK + wave;
    const unsigned m0   = tile * 16u;         // this wave's 16 query rows

    // ---- Load A operand (16x4 f32, K=3 valid): lanes0-15:(x,y)  lanes16-31:(z,0)
    const float* q = S1 + ((size_t)b * NPTS + (m0 + col)) * 3u;
    v2f a;
    if (!hi) { a[0] = q[0]; a[1] = q[1]; }
    else     { a[0] = q[2]; a[1] = 0.0f; }

    float best[8];
    int   bidx[8];
#pragma unroll
    for (int i = 0; i < 8; ++i) { best[i] = 3.4e38f; bidx[i] = 0; }

    const float* s2b = S2 + (size_t)b * NPTS * 3u;
    const v4f*   s2v = (const v4f*)s2b;       // NPTS*3 floats divisible by 4

    for (unsigned cbase = 0; cbase < NPTS; cbase += CHUNK_PTS) {
        __syncthreads();
        // ---- Stage chunk AoS -> SoA LDS with coalesced b128 loads
        const unsigned f4base = (cbase * 3u) >> 2;   // float4 offset of chunk
#pragma unroll
        for (unsigned k = 0; k < (CHUNK_PTS * 3u / 4u) / BLK_THREADS; ++k) {
            unsigned v4i = tid + k * BLK_THREADS;
            v4f v = s2v[f4base + v4i];
            unsigned fbase = v4i << 2;               // float index within chunk
#pragma unroll
            for (unsigned c = 0; c < 4u; ++c) {
                unsigned f    = fbase + c;
                unsigned p    = f / 3u;
                unsigned comp = f - p * 3u;
                sP[comp * CHUNK_PTS + p] = v[c];
            }
        }
        __syncthreads();

        // ---- Sweep 16-candidate tiles of this chunk
#pragma unroll 2
        for (unsigned t = 0; t < CHUNK_PTS / 16u; ++t) {
            unsigned nl = t * 16u + col;             // local candidate index
            float px = sP[nl];
            float py = sP[CHUNK_PTS + nl];
            float pz = sP[2u * CHUNK_PTS + nl];
            float bn2 = __builtin_fmaf(px, px,
                         __builtin_fmaf(py, py, pz * pz));   // ||c||^2

            // B operand (4x16 f32): lanes0-15:(x,y)  lanes16-31:(z,0)
            v2f bv;
            if (!hi) { bv[0] = px; bv[1] = py; }
            else     { bv[0] = pz; bv[1] = 0.0f; }

            v8f c = {};  // D = A x B : 16x16 tile of q.c dot products
            c = __builtin_amdgcn_wmma_f32_16x16x4_f32(
                    /*neg_a=*/false, a, /*neg_b=*/false, bv,
                    /*c_mod=*/(short)0, c, /*reuse_a=*/false, /*reuse_b=*/false);

            int n = (int)(cbase + nl);
#pragma unroll
            for (int i = 0; i < 8; ++i) {
                float score = __builtin_fmaf(-2.0f, c[i], bn2); // ||c||^2 - 2 q.c
                if (score < best[i]) { best[i] = score; bidx[i] = n; }
            }
        }
    }

    // ---- Min/argmin reduction across the 16 lanes of each half-wave
#pragma unroll
    for (int i = 0; i < 8; ++i) {
        float bs = best[i];
        int   bi = bidx[i];
#pragma unroll
        for (int off = 1; off < 16; off <<= 1) {
            float os = __shfl_xor(bs, off, 32);
            int   oi = __shfl_xor(bi, off, 32);
            if (os < bs || (os == bs && oi < bi)) { bs = os; bi = oi; }
        }
        best[i] = bs; bidx[i] = bi;
    }

    // lanes 0..15 hold rows m0+0..7 (c[0..7]); lanes 16..31 hold rows m0+8..15
    if (col == 0u) {
        unsigned rowbase = m0 + (hi ? 8u : 0u);
        float* o = out + (size_t)b * NPTS + rowbase;
#pragma unroll
        for (int i = 0; i < 8; ++i) o[i] = (float)bidx[i];
    }
}

extern "C" void kernel_launch(void* const* d_in, const int* in_sizes, int n_in,
                              void* d_out, int out_size, void* d_ws, size_t ws_size,
                              hipStream_t stream) {
    (void)in_sizes; (void)n_in; (void)out_size; (void)d_ws; (void)ws_size;
    const float* S1 = (const float*)d_in[0];
    const float* S2 = (const float*)d_in[1];
    float* out = (float*)d_out;

    const unsigned blocks = 2u * (NPTS / 16u) / WAVES_PER_BLK; // 256
    sided_nn_wmma_kernel<<<blocks, BLK_THREADS, 0, stream>>>(S1, S2, out);
}